// AdditiveAttn_11716670784117
// MI455X (gfx1250) — compile-verified
//
#include <hip/hip_runtime.h>
#include <hip/hip_bf16.h>

#define B_    64
#define S_    4096
#define A_    128
#define FDIM_ 512
#define HDIM_ 512
#define SSPLIT 32

typedef __attribute__((ext_vector_type(2))) float v2f;
typedef __attribute__((ext_vector_type(4))) float v4f;
typedef __attribute__((ext_vector_type(8))) float v8f;

// ---------------------------------------------------------------------------
// Kernel 1: proj_hidden[B,A] = hidden_state[B,HDIM] @ W_hidden[A,HDIM]^T
// One wave per 16x16 output tile, K-loop of V_WMMA_F32_16X16X4_F32.
// A-frag (16x4): lanes 0-15 hold K=0(K=1 in v1), lanes 16-31 hold K=2(K=3).
// B-frag (4x16): lane n holds column n; lanes 0-15 rows K0/K1, 16-31 rows K2/K3.
// Both source matrices are row-major over HDIM, so the per-lane load pattern
// is identical for A and B fragments.
// ---------------------------------------------------------------------------
__global__ __launch_bounds__(32) void proj_wmma_kernel(
    const float* __restrict__ hs,   // [B, HDIM]
    const float* __restrict__ Wh,   // [A, HDIM]
    float* __restrict__ ph)         // [B, A]
{
    const int tile = blockIdx.x;        // 0..31
    const int bt   = tile >> 3;         // batch tile 0..3
    const int at   = tile & 7;          // attn  tile 0..7
    const int lane = threadIdx.x;       // 0..31 (wave32)
    const int half = lane >> 4;         // 0: K0/K1, 1: K2/K3
    const int lid  = lane & 15;

    const float* hrow = hs + (size_t)(bt * 16 + lid) * HDIM_;
    const float* wrow = Wh + (size_t)(at * 16 + lid) * HDIM_;

    v8f c = {0.f, 0.f, 0.f, 0.f, 0.f, 0.f, 0.f, 0.f};

    #pragma unroll 4
    for (int k = 0; k < HDIM_; k += 4) {
        const int ko = k + 2 * half;
        v2f a, b;
        a.x = hrow[ko + 0];
        a.y = hrow[ko + 1];
        b.x = wrow[ko + 0];
        b.y = wrow[ko + 1];
        // 8 args: (neg_a, A, neg_b, B, c_mod, C, reuse_a, reuse_b)
        c = __builtin_amdgcn_wmma_f32_16x16x4_f32(
                false, a, false, b, (short)0, c, false, false);
    }

    // C/D layout: VGPR r, lanes 0-15 -> M=r, lanes 16-31 -> M=r+8; N=lid.
    float* out = ph + (size_t)(bt * 16 + half * 8) * A_ + at * 16 + lid;
    #pragma unroll
    for (int r = 0; r < 8; ++r)
        out[(size_t)r * A_] = c[r];
}

// ---------------------------------------------------------------------------
// Kernel 2: scores[b,s] = sum_a tanh(proj_image[b,s,a] + proj_hidden[b,a]) * w[a]
// One wave per (b,s) row: 32 lanes x v4f = 512B contiguous NT load.
// ---------------------------------------------------------------------------
__global__ __launch_bounds__(256) void scores_kernel(
    const float* __restrict__ pi,     // [B,S,A]
    const float* __restrict__ ph,     // [B,A]
    const float* __restrict__ wsc,    // [A]
    float* __restrict__ scores)       // [B,S]
{
    const int lane = threadIdx.x & 31;
    const int wave = threadIdx.x >> 5;
    const long long idx = (long long)blockIdx.x * 8 + wave;   // flat (b*S + s)
    const int b = (int)(idx >> 12);                            // S = 4096

    v4f p = __builtin_nontemporal_load((const v4f*)(pi + idx * A_) + lane);
    v4f h = *((const v4f*)(ph + (size_t)b * A_) + lane);
    v4f w = *((const v4f*)wsc + lane);

    float s = tanhf(p.x + h.x) * w.x
            + tanhf(p.y + h.y) * w.y
            + tanhf(p.z + h.z) * w.z
            + tanhf(p.w + h.w) * w.w;

    #pragma unroll
    for (int off = 16; off > 0; off >>= 1)
        s += __shfl_xor(s, off, 32);

    if (lane == 0)
        scores[idx] = s;
}

// ---------------------------------------------------------------------------
// Kernel 3: softmax over S per batch row. One block per b.
// ---------------------------------------------------------------------------
__global__ __launch_bounds__(256) void softmax_kernel(
    const float* __restrict__ scores,  // [B,S]
    float* __restrict__ weights)       // [B,S] (d_out weights region)
{
    __shared__ float red[256];
    const int b   = blockIdx.x;
    const int tid = threadIdx.x;
    const float* srow = scores + (size_t)b * S_;

    float m = -__builtin_inff();
    for (int s = tid; s < S_; s += 256) m = fmaxf(m, srow[s]);
    red[tid] = m; __syncthreads();
    for (int st = 128; st > 0; st >>= 1) {
        if (tid < st) red[tid] = fmaxf(red[tid], red[tid + st]);
        __syncthreads();
    }
    m = red[0]; __syncthreads();

    float sum = 0.f;
    for (int s = tid; s < S_; s += 256) sum += __expf(srow[s] - m);
    red[tid] = sum; __syncthreads();
    for (int st = 128; st > 0; st >>= 1) {
        if (tid < st) red[tid] += red[tid + st];
        __syncthreads();
    }
    const float inv = 1.f / red[0];

    for (int s = tid; s < S_; s += 256)
        weights[(size_t)b * S_ + s] = __expf(srow[s] - m) * inv;
}

// ---------------------------------------------------------------------------
// Kernel 4: zero the context output region (d_out is poisoned by harness).
// ---------------------------------------------------------------------------
__global__ __launch_bounds__(256) void zero_ctx_kernel(float* __restrict__ ctx)
{
    ctx[blockIdx.x * 256 + threadIdx.x] = 0.f;
}

// ---------------------------------------------------------------------------
// Kernel 5: context[b,d] = sum_s weights[b,s] * feat[b,s,d]
// Grid = B * SSPLIT; 128 threads x v4f cover FDIM=512 exactly.
// Each s-iteration reads one full 2KB feature row (coalesced b128 NT loads).
// Partials combined with hardware f32 global atomics.
// ---------------------------------------------------------------------------
__global__ __launch_bounds__(128) void context_kernel(
    const float* __restrict__ feat,    // [B,S,FDIM]
    const float* __restrict__ weights, // [B,S]
    float* __restrict__ ctx)           // [B,FDIM]
{
    const int sc = blockIdx.x % SSPLIT;
    const int b  = blockIdx.x / SSPLIT;
    const int d  = threadIdx.x * 4;
    const int SCHUNK = S_ / SSPLIT;    // 128

    const float* frow = feat + ((long long)b * S_ + (long long)sc * SCHUNK) * FDIM_ + d;
    const float* wrow = weights + (size_t)b * S_ + sc * SCHUNK;

    v4f acc = {0.f, 0.f, 0.f, 0.f};
    for (int s = 0; s < SCHUNK; ++s) {
        const float w = wrow[s];
        v4f f = __builtin_nontemporal_load((const v4f*)(frow + (long long)s * FDIM_));
        acc.x = fmaf(w, f.x, acc.x);
        acc.y = fmaf(w, f.y, acc.y);
        acc.z = fmaf(w, f.z, acc.z);
        acc.w = fmaf(w, f.w, acc.w);
    }

    float* o = ctx + (size_t)b * FDIM_ + d;
    __hip_atomic_fetch_add(o + 0, acc.x, __ATOMIC_RELAXED, __HIP_MEMORY_SCOPE_AGENT);
    __hip_atomic_fetch_add(o + 1, acc.y, __ATOMIC_RELAXED, __HIP_MEMORY_SCOPE_AGENT);
    __hip_atomic_fetch_add(o + 2, acc.z, __ATOMIC_RELAXED, __HIP_MEMORY_SCOPE_AGENT);
    __hip_atomic_fetch_add(o + 3, acc.w, __ATOMIC_RELAXED, __HIP_MEMORY_SCOPE_AGENT);
}

// ---------------------------------------------------------------------------
extern "C" void kernel_launch(void* const* d_in, const int* in_sizes, int n_in,
                              void* d_out, int out_size, void* d_ws, size_t ws_size,
                              hipStream_t stream) {
    const float* proj_image = (const float*)d_in[0];   // [B,S,A]
    const float* image_feat = (const float*)d_in[1];   // [B,S,FDIM]
    const float* hidden     = (const float*)d_in[2];   // [B,HDIM]
    const float* W_hidden   = (const float*)d_in[3];   // [A,HDIM]
    const float* w_score    = (const float*)d_in[4];   // [A]

    float* out_ctx = (float*)d_out;                    // [B,FDIM]
    float* out_w   = out_ctx + (size_t)B_ * FDIM_;     // [B,S]

    float* ws_ph     = (float*)d_ws;                   // [B,A]
    float* ws_scores = ws_ph + (size_t)B_ * A_;        // [B,S]

    // 1) proj_hidden GEMM via WMMA: 32 tiles of 16x16
    proj_wmma_kernel<<<32, 32, 0, stream>>>(hidden, W_hidden, ws_ph);

    // 2) scores: one wave per (b,s); 8 waves per block
    scores_kernel<<<(B_ * S_) / 8, 256, 0, stream>>>(proj_image, ws_ph, w_score, ws_scores);

    // 3) softmax per batch row -> weights output region
    softmax_kernel<<<B_, 256, 0, stream>>>(ws_scores, out_w);

    // 4) zero context accumulator
    zero_ctx_kernel<<<(B_ * FDIM_) / 256, 256, 0, stream>>>(out_ctx);

    // 5) weighted feature sum -> context
    context_kernel<<<B_ * SSPLIT, 128, 0, stream>>>(image_feat, out_w, out_ctx);
}